// MultiHeadGlobalAttention_81140522156282
// MI455X (gfx1250) — compile-verified
//
#include <hip/hip_runtime.h>
#include <cmath>

typedef __attribute__((ext_vector_type(2))) float v2f;
typedef __attribute__((ext_vector_type(8))) float v8f;

#define DIM    1024
#define SEQ    8192
#define BATCH  8
#define NCHUNK 64          // s-chunks for fused pass
#define ROWS   128         // rows per chunk (ROWS * NCHUNK == SEQ)
#define TILE   32          // rows staged in LDS per sub-tile (128 KB)

// workspace layout (float offsets)
#define WS_QT     0        // qT   [DIM][16]                 16384
#define WS_QTP    16384    // qt   [8][DIM]                   8192
#define WS_PARTM  24576    // m_c  [8][NCHUNK]                 512
#define WS_PARTP  25088    // p_c  [8][NCHUNK]                 512
#define WS_BSTAT  25600    // {M_b, 1/S_b} per batch            32
#define WS_CTXT   25632    // ctxT [DIM][16]                 16384
#define WS_PART   42016    // partials [8][NCHUNK][DIM]     524288
// total 566304 floats ~= 2.16 MB

// ---------------------------------------------------------------------------
// Mini-GEMMs: D(16x16 f32) += A(16x4) * B(4x16) via v_wmma_f32_16x16x4_f32.
// 8 blocks x 256 threads = 64 waves; one wave per 16-row M tile, K-loop of 256.
// Lane roles (ISA 7.12.2 f32 layouts): nn = lane&15 is the A row within the
// tile AND the B column; kk = (lane<16 ? 0 : 2); A/B VGPR j holds K = kk+j.
// C/D: vgpr r, lane -> M = r + 8*(lane>=16), N = nn.
// ---------------------------------------------------------------------------

// qT[e][16] = Wq(row-major) x x_last^T  (cols >= 8 clamped to b=7)
__global__ void k_qproj(const float* __restrict__ x, const float* __restrict__ Wq,
                        float* __restrict__ ws) {
    float* qT = ws + WS_QT;
    const int wave = threadIdx.x >> 5;
    const int lane = threadIdx.x & 31;
    const int m0   = (blockIdx.x * 8 + wave) * 16;
    const int nn   = lane & 15;
    const int kk   = (lane < 16) ? 0 : 2;
    const int bcol = (nn < BATCH) ? nn : (BATCH - 1);
    const float* arow  = Wq + (size_t)(m0 + nn) * DIM;
    const float* bcolp = x + ((size_t)bcol * SEQ + (SEQ - 1)) * DIM;
    v8f c = {};
#pragma unroll 4
    for (int k = 0; k < DIM; k += 4) {
        v2f a, b;
        a.x = arow[k + kk];   a.y = arow[k + kk + 1];
        b.x = bcolp[k + kk];  b.y = bcolp[k + kk + 1];
        c = __builtin_amdgcn_wmma_f32_16x16x4_f32(false, a, false, b,
                                                  (short)0, c, false, false);
    }
    const int mhi = (lane < 16) ? 0 : 8;
#pragma unroll
    for (int r = 0; r < 8; ++r)
        qT[(size_t)(m0 + r + mhi) * 16 + nn] = c[r];
}

// qt[b][d] : qt^T[d,b] = sum_e Wk[e,d] * qT[e,b]
__global__ void k_qk(const float* __restrict__ Wk, float* __restrict__ ws) {
    const float* qT = ws + WS_QT;
    float* qt = ws + WS_QTP;
    const int wave = threadIdx.x >> 5;
    const int lane = threadIdx.x & 31;
    const int m0   = (blockIdx.x * 8 + wave) * 16;
    const int nn   = lane & 15;
    const int kk   = (lane < 16) ? 0 : 2;
    const int mrow = m0 + nn;
    v8f c = {};
#pragma unroll 4
    for (int k = 0; k < DIM; k += 4) {
        v2f a, b;
        a.x = Wk[(size_t)(k + kk) * DIM + mrow];
        a.y = Wk[(size_t)(k + kk + 1) * DIM + mrow];
        b.x = qT[(size_t)(k + kk) * 16 + nn];
        b.y = qT[(size_t)(k + kk + 1) * 16 + nn];
        c = __builtin_amdgcn_wmma_f32_16x16x4_f32(false, a, false, b,
                                                  (short)0, c, false, false);
    }
    const int mhi = (lane < 16) ? 0 : 8;
    if (nn < BATCH) {
#pragma unroll
        for (int r = 0; r < 8; ++r)
            qt[(size_t)nn * DIM + (m0 + r + mhi)] = c[r];
    }
}

// context^T[d',b] = sum_d Wv[d',d] * ctxT[d,b] ; scatter to out[b][d']
__global__ void k_outproj(const float* __restrict__ Wv, const float* __restrict__ ws,
                          float* __restrict__ out) {
    const float* ctxT = ws + WS_CTXT;
    const int wave = threadIdx.x >> 5;
    const int lane = threadIdx.x & 31;
    const int m0   = (blockIdx.x * 8 + wave) * 16;
    const int nn   = lane & 15;
    const int kk   = (lane < 16) ? 0 : 2;
    const float* arow = Wv + (size_t)(m0 + nn) * DIM;
    v8f c = {};
#pragma unroll 4
    for (int k = 0; k < DIM; k += 4) {
        v2f a, b;
        a.x = arow[k + kk];  a.y = arow[k + kk + 1];
        b.x = ctxT[(size_t)(k + kk) * 16 + nn];
        b.y = ctxT[(size_t)(k + kk + 1) * 16 + nn];
        c = __builtin_amdgcn_wmma_f32_16x16x4_f32(false, a, false, b,
                                                  (short)0, c, false, false);
    }
    const int mhi = (lane < 16) ? 0 : 8;
    if (nn < BATCH) {
#pragma unroll
        for (int r = 0; r < 8; ++r)
            out[(size_t)nn * DIM + (m0 + r + mhi)] = c[r];
    }
}

// ---------------------------------------------------------------------------
// Fused single-HBM-pass with LDS tile staging + online softmax.
// Per (b, chunk of ROWS rows), in sub-tiles of TILE rows:
//   A) stream rows HBM -> VGPR: dot with qt (wave reduce) AND ds_store to LDS
//   B) wave0: tile max + weights; all threads: online rescale (m_run, p_run, acc)
//   C) weighted accumulation reading the tile from LDS (no second HBM touch)
// x is read from HBM exactly once. Partials combined deterministically after.
// ---------------------------------------------------------------------------
__global__ void k_fused(const float* __restrict__ x, float* __restrict__ ws) {
    extern __shared__ float smem[];          // [DIM] qts | [TILE*DIM] xtile
    float* qts   = smem;
    float* xtile = smem + DIM;
    __shared__ float sc[TILE];
    __shared__ float wts[TILE];
    __shared__ float bc[2];

    const int b = blockIdx.y, chunk = blockIdx.x;
    const int s0 = chunk * ROWS;
    const int tid = threadIdx.x, lane = tid & 31, wave = tid >> 5;

    const float* qt = ws + WS_QTP + (size_t)b * DIM;
    for (int i = tid; i < DIM / 4; i += 256)
        ((float4*)qts)[i] = ((const float4*)qt)[i];

    const int d = tid * 4;
    float4 acc = make_float4(0.f, 0.f, 0.f, 0.f);
    float m_run = -3.4e38f, p_run = 0.f;
    __syncthreads();

    for (int t = 0; t < ROWS / TILE; ++t) {
        const int sbase = s0 + t * TILE;
        // --- A: load tile rows, stage to LDS, compute dots ---
        for (int i = 0; i < TILE / 8; ++i) {
            const int r = i * 8 + wave;
            const float* xr = x + ((size_t)b * SEQ + sbase + r) * DIM;
            float dot = 0.f;
#pragma unroll
            for (int j = 0; j < 8; ++j) {
                const int dd = j * 128 + lane * 4;
                float4 xv = *(const float4*)(xr + dd);
                *(float4*)(xtile + (size_t)r * DIM + dd) = xv;
                float4 qv = *(const float4*)(qts + dd);
                dot = fmaf(xv.x, qv.x, dot);
                dot = fmaf(xv.y, qv.y, dot);
                dot = fmaf(xv.z, qv.z, dot);
                dot = fmaf(xv.w, qv.w, dot);
            }
            for (int off = 16; off; off >>= 1)
                dot += __shfl_xor(dot, off, 32);
            if (lane == 0) sc[r] = dot * 0.03125f;      // 1/sqrt(1024)
        }
        __syncthreads();

        // --- B: tile stats (wave0) + online rescale (all threads) ---
        if (wave == 0) {
            const float sv = sc[lane];                   // TILE == 32
            float tm = sv;
            for (int off = 16; off; off >>= 1)
                tm = fmaxf(tm, __shfl_xor(tm, off, 32));
            const float nm = fmaxf(m_run, tm);
            float wv = __expf(sv - nm);
            wts[lane] = wv;
            for (int off = 16; off; off >>= 1)
                wv += __shfl_xor(wv, off, 32);
            if (lane == 0) { bc[0] = nm; bc[1] = wv; }
        }
        __syncthreads();
        const float nm   = bc[0];
        const float tp   = bc[1];
        const float corr = __expf(m_run - nm);           // first tile: -> 0
        p_run = p_run * corr + tp;
        m_run = nm;
        acc.x *= corr; acc.y *= corr; acc.z *= corr; acc.w *= corr;

        // --- C: weighted accumulation from LDS ---
        for (int i = 0; i < TILE; ++i) {
            const float wi = wts[i];
            float4 xv = *(const float4*)(xtile + (size_t)i * DIM + d);
            acc.x = fmaf(wi, xv.x, acc.x);
            acc.y = fmaf(wi, xv.y, acc.y);
            acc.z = fmaf(wi, xv.z, acc.z);
            acc.w = fmaf(wi, xv.w, acc.w);
        }
        __syncthreads();   // tile consumed before next overwrite
    }

    *(float4*)(ws + WS_PART + ((size_t)(b * NCHUNK + chunk)) * DIM + d) = acc;
    if (tid == 0) {
        ws[WS_PARTM + (size_t)b * NCHUNK + chunk] = m_run;   // == chunk max
        ws[WS_PARTP + (size_t)b * NCHUNK + chunk] = p_run;
    }
}

// Per-batch global stats: M_b = max_c m_c ; S_b = sum_c p_c * exp(m_c - M_b)
// One wave per batch; lane handles chunks {lane, lane+32}. Butterfly reduce.
__global__ void k_bstats(float* __restrict__ ws) {
    const int b = blockIdx.x;
    const int lane = threadIdx.x;
    const float* pm = ws + WS_PARTM + (size_t)b * NCHUNK;
    const float* pp = ws + WS_PARTP + (size_t)b * NCHUNK;
    float m0 = pm[lane], m1 = pm[lane + 32];
    float m = fmaxf(m0, m1);
    for (int off = 16; off; off >>= 1) m = fmaxf(m, __shfl_xor(m, off, 32));
    float s = pp[lane] * __expf(m0 - m) + pp[lane + 32] * __expf(m1 - m);
    for (int off = 16; off; off >>= 1) s += __shfl_xor(s, off, 32);
    if (lane == 0) {
        ws[WS_BSTAT + 2 * b]     = m;
        ws[WS_BSTAT + 2 * b + 1] = 1.f / s;
    }
}

// Deterministic fixed-order combine: ctxT[d][b] = invS_b * sum_c exp(m_c-M_b)*part_c[d]
__global__ void k_ctx_combine(float* __restrict__ ws) {
    const int gid = blockIdx.x * 256 + threadIdx.x;   // 0..8191
    const int b = gid >> 10;
    const int d = gid & 1023;
    const float Mb   = ws[WS_BSTAT + 2 * b];
    const float invS = ws[WS_BSTAT + 2 * b + 1];
    const float* pm   = ws + WS_PARTM + (size_t)b * NCHUNK;
    const float* part = ws + WS_PART + (size_t)b * NCHUNK * DIM + d;
    float s = 0.f;
    for (int c = 0; c < NCHUNK; ++c)
        s = fmaf(__expf(pm[c] - Mb), part[(size_t)c * DIM], s);
    float* ctxT = ws + WS_CTXT;
    ctxT[(size_t)d * 16 + b] = s * invS;
    if (b == 0) {
        for (int n = BATCH; n < 16; ++n) ctxT[(size_t)d * 16 + n] = 0.f;
    }
}

extern "C" void kernel_launch(void* const* d_in, const int* in_sizes, int n_in,
                              void* d_out, int out_size, void* d_ws, size_t ws_size,
                              hipStream_t stream) {
    const float* x  = (const float*)d_in[0];
    const float* Wq = (const float*)d_in[1];
    const float* Wk = (const float*)d_in[2];
    const float* Wv = (const float*)d_in[3];
    float* out = (float*)d_out;
    float* ws  = (float*)d_ws;
    (void)in_sizes; (void)n_in; (void)out_size; (void)ws_size;

    const size_t fused_lds = (size_t)(DIM + TILE * DIM) * sizeof(float); // 132 KB

    k_qproj      <<<8, 256, 0, stream>>>(x, Wq, ws);
    k_qk         <<<8, 256, 0, stream>>>(Wk, ws);
    k_fused      <<<dim3(NCHUNK, BATCH), 256, fused_lds, stream>>>(x, ws);
    k_bstats     <<<BATCH, 32, 0, stream>>>(ws);
    k_ctx_combine<<<(BATCH * DIM) / 256, 256, 0, stream>>>(ws);
    k_outproj    <<<8, 256, 0, stream>>>(Wv, ws, out);
}